// GIN_60163901882503
// MI455X (gfx1250) — compile-verified
//
#include <hip/hip_runtime.h>
#include <hip/hip_bf16.h>

typedef __attribute__((ext_vector_type(2))) float v2f;
typedef __attribute__((ext_vector_type(8))) float v8f;

#define BN_EPS 1e-5f

// ---------------------------------------------------------------------------
// Zero fill (grid-stride)
// ---------------------------------------------------------------------------
__global__ __launch_bounds__(256) void gin_zero(float* __restrict__ p, long long n) {
    long long i = (long long)blockIdx.x * 256 + threadIdx.x;
    long long stride = (long long)gridDim.x * 256;
    for (; i < n; i += stride) p[i] = 0.0f;
}

// ---------------------------------------------------------------------------
// Edge scatter: agg[dst[e]] += x[src[e]]   (64 f32 per edge, float4 per lane)
// 16 consecutive lanes cover one full 256B feature row -> coalesced gathers;
// accumulation via no-return global_atomic_add_f32 (resolves in L2).
// ---------------------------------------------------------------------------
__global__ __launch_bounds__(256) void gin_scatter(
    const float* __restrict__ X, const int* __restrict__ src,
    const int* __restrict__ dst, float* __restrict__ A, int E) {
    long long gid = (long long)blockIdx.x * 256 + threadIdx.x;
    long long tot = (long long)E * 16;
    if (gid >= tot) return;
    long long e = gid >> 4;
    int q = (int)(gid & 15) * 4;
    long long s = (long long)src[e];
    long long d = (long long)dst[e];
    float4 v = *(const float4*)(X + s * 64 + q);
    float* p = A + d * 64 + q;
    atomicAdd(p + 0, v.x);
    atomicAdd(p + 1, v.y);
    atomicAdd(p + 2, v.z);
    atomicAdd(p + 3, v.w);
}

// ---------------------------------------------------------------------------
// Fused GIN linear: Z = ((1+eps)*X + AGG) @ W + bias      using v_wmma_f32_16x16x4_f32
// Block = 256 threads (8 wave32). Block computes 128 rows x NC cols.
// Each wave owns a 16-row strip and loops over NC/16 column tiles; K=64 -> 16 WMMA.
// Optionally accumulates per-column sum / sum-of-squares for the following BatchNorm.
// LDS A-tile padded to stride 66 -> conflict-free, 8B-aligned float2 fragment reads.
// ---------------------------------------------------------------------------
template<int NC, bool FUSE, bool STATS>
__global__ __launch_bounds__(256) void gin_gemm(
    const float* __restrict__ X, const float* __restrict__ AGG,
    const float* __restrict__ epsp,
    const float* __restrict__ W, const float* __restrict__ bias,
    float* __restrict__ Zout, float* __restrict__ statS, float* __restrict__ statQ,
    int N) {
    __shared__ float lh[128 * 66];
    __shared__ float lw[64 * NC];
    __shared__ float ls[NC];
    __shared__ float lq[NC];

    const int t = threadIdx.x;
    const long long rowBase = (long long)blockIdx.x * 128;
    const float scale = FUSE ? (1.0f + epsp[0]) : 1.0f;

    // Stage 128x64 input tile: h = scale*X + AGG (coalesced float4 loads)
    #pragma unroll
    for (int i = 0; i < 8; ++i) {
        int lin = i * 1024 + t * 4;
        int r = lin >> 6;
        int c = lin & 63;
        long long gr = rowBase + r;
        float4 v = make_float4(0.f, 0.f, 0.f, 0.f);
        if (gr < N) {
            v = *(const float4*)(X + gr * 64 + c);
            if (FUSE) {
                float4 a = *(const float4*)(AGG + gr * 64 + c);
                v.x = scale * v.x + a.x;
                v.y = scale * v.y + a.y;
                v.z = scale * v.z + a.z;
                v.w = scale * v.w + a.w;
            }
        }
        float* dptr = &lh[r * 66 + c];
        dptr[0] = v.x; dptr[1] = v.y; dptr[2] = v.z; dptr[3] = v.w;
    }
    // Stage W (64 x NC)
    for (int lin = t; lin < 64 * NC; lin += 256) lw[lin] = W[lin];
    if (STATS && t < NC) { ls[t] = 0.f; lq[t] = 0.f; }
    __syncthreads();

    const int w    = t >> 5;
    const int lane = t & 31;
    const int lm   = lane & 15;
    const int hi   = lane >> 4;
    const int rloc = w * 16;

    for (int ct = 0; ct < NC / 16; ++ct) {
        const int col0 = ct * 16;
        v8f acc = {};
        #pragma unroll
        for (int s = 0; s < 16; ++s) {
            const int k = 4 * s + 2 * hi;
            // A frag (16x4): lane<16 -> K={k0,k0+1}, lane>=16 -> K={k0+2,k0+3}
            v2f a;
            a.x = lh[(rloc + lm) * 66 + k];
            a.y = lh[(rloc + lm) * 66 + k + 1];
            // B frag (4x16): same K split, N striped across lanes
            v2f b;
            b.x = lw[k * NC + col0 + lm];
            b.y = lw[(k + 1) * NC + col0 + lm];
            acc = __builtin_amdgcn_wmma_f32_16x16x4_f32(
                false, a, false, b, (short)0, acc, false, false);
        }
        // C/D layout: VGPR i = row (i + 8*hi), col = col0 + lm
        const int c = col0 + lm;
        const float bv = bias[c];
        float sSum = 0.f, sSq = 0.f;
        #pragma unroll
        for (int i = 0; i < 8; ++i) {
            long long gr = rowBase + rloc + hi * 8 + i;
            if (gr < N) {
                float v = acc[i] + bv;
                Zout[gr * NC + c] = v;
                if (STATS) { sSum += v; sSq += v * v; }
            }
        }
        if (STATS) {
            atomicAdd(&ls[c], sSum);
            atomicAdd(&lq[c], sSq);
        }
    }
    if (STATS) {
        __syncthreads();
        if (t < NC) {
            atomicAdd(&statS[t], ls[t]);
            atomicAdd(&statQ[t], lq[t]);
        }
    }
}

// ---------------------------------------------------------------------------
// BatchNorm apply (+optional ReLU) from accumulated column sums, optionally
// accumulating stats of the produced tensor for the next BatchNorm.
// ---------------------------------------------------------------------------
template<bool RELU, bool STATS>
__global__ __launch_bounds__(256) void gin_bn(
    const float* __restrict__ Z, const float* __restrict__ S, const float* __restrict__ Q,
    const float* __restrict__ g, const float* __restrict__ be,
    float* __restrict__ Y, float* __restrict__ oS, float* __restrict__ oQ, int N) {
    __shared__ float ls[64];
    __shared__ float lq[64];
    const int t = threadIdx.x;
    if (STATS) {
        if (t < 64) { ls[t] = 0.f; lq[t] = 0.f; }
        __syncthreads();
    }
    long long idx = (long long)blockIdx.x * 256 + t;
    long long tot = (long long)N * 64;
    float y = 0.f;
    int c = 0;
    bool act = idx < tot;
    if (act) {
        c = (int)(idx & 63);
        const float inv = 1.0f / (float)N;
        const float mean = S[c] * inv;
        const float var = Q[c] * inv - mean * mean;
        const float rs = rsqrtf(var + BN_EPS);
        float z = Z[idx];
        y = (z - mean) * rs * g[c] + be[c];
        if (RELU) y = fmaxf(y, 0.f);
        Y[idx] = y;
    }
    if (STATS) {
        if (act) {
            atomicAdd(&ls[c], y);
            atomicAdd(&lq[c], y * y);
        }
        __syncthreads();
        if (t < 64) {
            atomicAdd(&oS[t], ls[t]);
            atomicAdd(&oQ[t], lq[t]);
        }
    }
}

// ---------------------------------------------------------------------------
// Host-side orchestration
// ---------------------------------------------------------------------------
extern "C" void kernel_launch(void* const* d_in, const int* in_sizes, int n_in,
                              void* d_out, int out_size, void* d_ws, size_t ws_size,
                              hipStream_t stream) {
    const float* x   = (const float*)d_in[0];
    const int*   ei  = (const int*)d_in[1];
    const float* e1  = (const float*)d_in[2];
    const float* W1  = (const float*)d_in[3];
    const float* b1  = (const float*)d_in[4];
    const float* g1m = (const float*)d_in[5];
    const float* be1m= (const float*)d_in[6];
    const float* g1  = (const float*)d_in[7];
    const float* be1 = (const float*)d_in[8];
    const float* e2  = (const float*)d_in[9];
    const float* W2  = (const float*)d_in[10];
    const float* b2  = (const float*)d_in[11];
    const float* g2m = (const float*)d_in[12];
    const float* be2m= (const float*)d_in[13];
    const float* g2  = (const float*)d_in[14];
    const float* be2 = (const float*)d_in[15];
    const float* Wp  = (const float*)d_in[16];
    const float* bp  = (const float*)d_in[17];

    const int N = in_sizes[0] / 64;
    const int E = in_sizes[1] / 2;
    const int* src = ei;
    const int* dst = ei + E;

    const long long N64 = (long long)N * 64;
    float* ws = (float*)d_ws;
    float* A  = ws;            // aggregation buffer [N,64]
    float* Z  = A + N64;       // pre-BN linear output [N,64]
    float* Y  = Z + N64;       // post-relu intermediate [N,64]
    float* H  = Y + N64;       // layer output [N,64]
    float* ST = H + N64;       // 8 x 64 stat accumulators
    float *s1 = ST,       *q1 = ST + 64,  *s2 = ST + 128, *q2 = ST + 192;
    float *s3 = ST + 256, *q3 = ST + 320, *s4 = ST + 384, *q4 = ST + 448;

    const dim3 blk(256);
    const int zgrid  = 2048;
    const int sgrid  = (int)(((long long)E * 16 + 255) / 256);
    const int ggrid  = (N + 127) / 128;
    const int bgrid  = (int)((N64 + 255) / 256);

    // ---- Layer 1 ----
    gin_zero<<<zgrid, blk, 0, stream>>>(A, N64);
    gin_zero<<<1, blk, 0, stream>>>(ST, 512);
    gin_scatter<<<sgrid, blk, 0, stream>>>(x, src, dst, A, E);
    gin_gemm<64, true, true><<<ggrid, blk, 0, stream>>>(x, A, e1, W1, b1, Z, s1, q1, N);
    gin_bn<true, true><<<bgrid, blk, 0, stream>>>(Z, s1, q1, g1m, be1m, Y, s2, q2, N);
    gin_bn<false, false><<<bgrid, blk, 0, stream>>>(Y, s2, q2, g1, be1, H, nullptr, nullptr, N);

    // ---- Layer 2 ----
    gin_zero<<<zgrid, blk, 0, stream>>>(A, N64);
    gin_scatter<<<sgrid, blk, 0, stream>>>(H, src, dst, A, E);
    gin_gemm<64, true, true><<<ggrid, blk, 0, stream>>>(H, A, e2, W2, b2, Z, s3, q3, N);
    gin_bn<true, true><<<bgrid, blk, 0, stream>>>(Z, s3, q3, g2m, be2m, Y, s4, q4, N);
    gin_bn<false, false><<<bgrid, blk, 0, stream>>>(Y, s4, q4, g2, be2, H, nullptr, nullptr, N);

    // ---- Output projection [N,64] @ [64,32] + bp ----
    gin_gemm<32, false, false><<<ggrid, blk, 0, stream>>>(H, nullptr, nullptr, Wp, bp,
                                                          (float*)d_out, nullptr, nullptr, N);
}